// GatedBlock_45097156608312
// MI455X (gfx1250) — compile-verified
//
#include <hip/hip_runtime.h>

#define FEAT     2560
#define SIZE_OUT 2112
#define N_GATES  448   // 256 (l=1) + 128 (l=2) + 64 (l=3)

typedef int v4i __attribute__((vector_size(16)));
typedef __attribute__((address_space(1))) v4i gas_v4i;  // global (AS1)
typedef __attribute__((address_space(3))) v4i las_v4i;  // LDS (AS3)

__device__ __forceinline__ float fast_sigmoid(float x) {
    return 1.0f / (1.0f + __expf(-x));
}

union F4 { float4 v; float f[4]; };

__global__ __launch_bounds__(256) void gated_block_kernel(
    const float* __restrict__ in, float* __restrict__ out, int n_rows) {

    __shared__ float sg[N_GATES];

    const int row = blockIdx.x;
    if (row >= n_rows) return;
    const int tid = threadIdx.x;

    const float* __restrict__ rowin  = in  + (size_t)row * FEAT;
    float* __restrict__       rowout = out + (size_t)row * SIZE_OUT;

    // ---- Phase 1: stage the 448 gates into LDS (CDNA5 async copy path) ----
#if __has_builtin(__builtin_amdgcn_global_load_async_to_lds_b128)
    if (tid < N_GATES / 4) {
        const float* gsrc = rowin + SIZE_OUT + tid * 4;
        __builtin_amdgcn_global_load_async_to_lds_b128(
            (gas_v4i*)gsrc,
            (las_v4i*)&sg[tid * 4],
            /*offset=*/0, /*cpol=*/0);
    }
  #if __has_builtin(__builtin_amdgcn_s_wait_asynccnt)
    __builtin_amdgcn_s_wait_asynccnt(0);
  #else
    asm volatile("s_wait_asynccnt 0" ::: "memory");
  #endif
#else
    if (tid < N_GATES / 4) {
        F4 g;
        g.v = *reinterpret_cast<const float4*>(rowin + SIZE_OUT + tid * 4);
        *reinterpret_cast<float4*>(&sg[tid * 4]) = g.v;
    }
#endif
    __syncthreads();

    // sigmoid once per gate, in place in LDS (448 gates, 256 threads)
    for (int i = tid; i < N_GATES; i += 256) {
        sg[i] = fast_sigmoid(sg[i]);
    }
    __syncthreads();

    // ---- Phase 2: stream 528 float4 chunks of the 2112-wide output ----
    // Region boundaries (256, 1024, 1664) are float4-aligned, so each chunk
    // lies entirely inside one region.
    for (int c = tid; c < SIZE_OUT / 4; c += 256) {
        const int col = c * 4;
        F4 x, y;
        x.v = *reinterpret_cast<const float4*>(rowin + col);

        if (col < 256) {                       // l=0: silu
            #pragma unroll
            for (int j = 0; j < 4; ++j)
                y.f[j] = x.f[j] * fast_sigmoid(x.f[j]);
        } else if (col < 1024) {               // l=1: gate idx (col-256)/3
            #pragma unroll
            for (int j = 0; j < 4; ++j)
                y.f[j] = x.f[j] * sg[(col + j - 256) / 3];
        } else if (col < 1664) {               // l=2: gate idx 256+(col-1024)/5
            #pragma unroll
            for (int j = 0; j < 4; ++j)
                y.f[j] = x.f[j] * sg[256 + (col + j - 1024) / 5];
        } else {                               // l=3: gate idx 384+(col-1664)/7
            #pragma unroll
            for (int j = 0; j < 4; ++j)
                y.f[j] = x.f[j] * sg[384 + (col + j - 1664) / 7];
        }

        *reinterpret_cast<float4*>(rowout + col) = y.v;
    }
}

extern "C" void kernel_launch(void* const* d_in, const int* in_sizes, int n_in,
                              void* d_out, int out_size, void* d_ws, size_t ws_size,
                              hipStream_t stream) {
    const float* in  = (const float*)d_in[0];
    float*       out = (float*)d_out;
    const int n_rows = in_sizes[0] / FEAT;

    gated_block_kernel<<<dim3(n_rows), dim3(256), 0, stream>>>(in, out, n_rows);
}